// Attention_nn_57707180589240
// MI455X (gfx1250) — compile-verified
//
#include <hip/hip_runtime.h>
#include <hip/hip_bf16.h>

typedef float v2f __attribute__((ext_vector_type(2)));
typedef float v4f __attribute__((ext_vector_type(4)));
typedef float v8f __attribute__((ext_vector_type(8)));

// Problem constants (from reference setup_inputs)
#define BATCH 256
#define HDE   512
#define EDIM  512
#define HW    256
#define CATT  512

// ---------------------------------------------------------------------------
// Kernel 1: g_em[b,e] = sum_k h[b,k] * Wh[e,k] + bh[e]
// FP32 WMMA 16x16x4. One wave (32 lanes) computes one 16x16 output tile.
// Block = 256 threads = 8 waves -> covers 16 (M) x 128 (N) per block.
// Grid = (256/16, 512/128) = (16, 4).
// ---------------------------------------------------------------------------
__global__ __launch_bounds__(256) void gemm_bias_wmma_f32(
    const float* __restrict__ h,    // [BATCH, HDE]
    const float* __restrict__ Wh,   // [EDIM, HDE]
    const float* __restrict__ bh,   // [EDIM]
    float*       __restrict__ g_em) // [BATCH, EDIM]
{
    const int lane = threadIdx.x & 31;
    const int wave = threadIdx.x >> 5;
    const int m0 = blockIdx.x * 16;                // batch-row tile
    const int n0 = blockIdx.y * 128 + wave * 16;   // E-column tile
    const int lm = lane & 15;
    const int khalf = (lane >> 4) << 1;            // lanes 0-15: K+0/K+1, lanes 16-31: K+2/K+3

    const float* arow = h  + (size_t)(m0 + lm) * HDE + khalf;
    const float* brow = Wh + (size_t)(n0 + lm) * HDE + khalf;  // B = Wh^T, per-lane row of Wh

    // Accumulator preloaded with bias (bias depends only on N = n0+lm).
    v8f c;
    const float bias = bh[n0 + lm];
#pragma unroll
    for (int r = 0; r < 8; ++r) c[r] = bias;

    for (int k = 0; k < HDE; k += 4) {
        v2f a = *(const v2f*)(arow + k);
        v2f b = *(const v2f*)(brow + k);
        // D = A(16x4,f32) x B(4x16,f32) + C(16x16,f32)
        c = __builtin_amdgcn_wmma_f32_16x16x4_f32(
                /*neg_a=*/false, a, /*neg_b=*/false, b,
                /*c_mod=*/(short)0, c, /*reuse_a=*/false, /*reuse_b=*/false);
    }

    // C/D layout: VGPR r -> row m0+r (lanes 0-15) / m0+8+r (lanes 16-31), col n0+(lane&15)
    const int moff = (lane >> 4) ? 8 : 0;
#pragma unroll
    for (int r = 0; r < 8; ++r) {
        g_em[(size_t)(m0 + moff + r) * EDIM + (n0 + lm)] = c[r];
    }
}

// ---------------------------------------------------------------------------
// Kernel 2 (fused, one block per batch row b, 512 threads):
//   scores[hw] = sum_e Wa[e]*tanh(x_em[b,e,hw] + g_em[b,e]) + ba
//   alpha      = softmax(scores)                 (written to d_out tail)
//   att_out[c] = sum_s alpha[s]*att_x[b,s,c]     (written to d_out head)
//
// Both 134 MB streams are read exactly once -> non-temporal b128 loads
// (float4 per lane = 512 B per wave per VMEM instruction) so the loop is
// bandwidth-limited, not VMEM-issue-limited, and doesn't pollute WGP$/L2.
// ---------------------------------------------------------------------------
__global__ __launch_bounds__(512) void attn_fused(
    const float* __restrict__ x_em,  // [BATCH, EDIM, HW]
    const float* __restrict__ att_x, // [BATCH, HW, CATT]
    const float* __restrict__ g_em,  // [BATCH, EDIM]
    const float* __restrict__ Wa,    // [EDIM]
    const float* __restrict__ ba_p,  // [1]
    float*       __restrict__ out)   // [BATCH*CATT] att_out, then [BATCH*HW] alpha
{
    __shared__ v4f   sred4[512];     // 8 KB partial-sum exchange
    __shared__ float sbuf[HW];       // softmax scratch
    __shared__ float salpha[HW];     // final attention weights

    const int b = blockIdx.x;
    const int t = threadIdx.x;

    // ---- stage 1: scores. thread owns hw4 = 4 consecutive hw; E split 8-way ----
    {
        const int tq = t & 63;       // hw block: [tq*4, tq*4+4)
        const int eh = t >> 6;       // e-slice: [eh*64, (eh+1)*64)
        const float* xb = x_em + (size_t)b * EDIM * HW + tq * 4;
        const float* gb = g_em + (size_t)b * EDIM;

        v4f acc = {0.f, 0.f, 0.f, 0.f};
#pragma unroll 4
        for (int e = eh * 64; e < (eh + 1) * 64; ++e) {
            v4f xv = __builtin_nontemporal_load((const v4f*)(xb + (size_t)e * HW));
            const float ge = gb[e];
            const float wa = Wa[e];
            v4f tv;
#pragma unroll
            for (int i = 0; i < 4; ++i) tv[i] = tanhf(xv[i] + ge);
            acc += wa * tv;
        }
        sred4[t] = acc;
    }
    __syncthreads();

    // combine the 8 e-slices -> 256 scores (64 threads x 4 hw each)
    if (t < 64) {
        v4f sc = sred4[t];
#pragma unroll
        for (int g = 1; g < 8; ++g) sc += sred4[g * 64 + t];
        const float ba = ba_p[0];
#pragma unroll
        for (int i = 0; i < 4; ++i) sbuf[t * 4 + i] = sc[i] + ba;
    }
    __syncthreads();

    // ---- softmax over HW=256 (LDS tree max, then sum) ----
    const float score = (t < HW) ? sbuf[t] : 0.f;
    for (int off = HW / 2; off > 0; off >>= 1) {
        if (t < off) sbuf[t] = fmaxf(sbuf[t], sbuf[t + off]);
        __syncthreads();
    }
    const float smax = sbuf[0];
    __syncthreads();

    float ex = 0.f;
    if (t < HW) { ex = __expf(score - smax); sbuf[t] = ex; }
    __syncthreads();
    for (int off = HW / 2; off > 0; off >>= 1) {
        if (t < off) sbuf[t] += sbuf[t + off];
        __syncthreads();
    }
    const float inv = 1.0f / sbuf[0];

    if (t < HW) {
        const float a = ex * inv;
        salpha[t] = a;
        // alpha output region starts after att_out (BATCH*CATT floats)
        out[(size_t)BATCH * CATT + (size_t)b * HW + t] = a;
    }
    __syncthreads();

    // ---- stage 2: att_out. thread owns c4 = 4 consecutive c; s split 4-way ----
    {
        const int cq = t & 127;      // c block: [cq*4, cq*4+4)
        const int sh = t >> 7;       // s-slice: [sh*64, (sh+1)*64)
        const float* ab = att_x + (size_t)b * HW * CATT + cq * 4;

        v4f o = {0.f, 0.f, 0.f, 0.f};
#pragma unroll 4
        for (int s = sh * 64; s < (sh + 1) * 64; ++s) {
            v4f xv = __builtin_nontemporal_load((const v4f*)(ab + (size_t)s * CATT));
            o += salpha[s] * xv;
        }
        sred4[t] = o;
    }
    __syncthreads();

    // combine the 4 s-slices and store att_out as b128 (128 threads x 4 c each)
    if (t < 128) {
        v4f o = sred4[t] + sred4[t + 128] + sred4[t + 256] + sred4[t + 384];
        *(v4f*)(out + (size_t)b * CATT + t * 4) = o;
    }
}

// ---------------------------------------------------------------------------
extern "C" void kernel_launch(void* const* d_in, const int* in_sizes, int n_in,
                              void* d_out, int out_size, void* d_ws, size_t ws_size,
                              hipStream_t stream) {
    const float* h     = (const float*)d_in[0];  // [256,512]
    const float* x_em  = (const float*)d_in[1];  // [256,512,8,32]
    const float* att_x = (const float*)d_in[2];  // [256,256,512]
    const float* Wh    = (const float*)d_in[3];  // [512,512]
    const float* bh    = (const float*)d_in[4];  // [512]
    const float* Wa    = (const float*)d_in[5];  // [512]
    const float* ba    = (const float*)d_in[6];  // [1]
    // d_in[7], d_in[8]: feature_h / feature_w scalars (hardcoded as 8, 32)

    float* out  = (float*)d_out;
    float* g_em = (float*)d_ws;                  // [256,512] = 512 KB scratch

    dim3 grid1(BATCH / 16, EDIM / 128);          // (16, 4), 256 threads = 8 waves
    gemm_bias_wmma_f32<<<grid1, 256, 0, stream>>>(h, Wh, bh, g_em);

    attn_fused<<<BATCH, 512, 0, stream>>>(x_em, att_x, g_em, Wa, ba, out);
}